// NetS3DIS_53068615909413
// MI455X (gfx1250) — compile-verified
//
#include <hip/hip_runtime.h>

typedef __attribute__((ext_vector_type(2))) float v2f;
typedef __attribute__((ext_vector_type(8))) float v8f;

#define PTS_STRIDE 8192   // all point sets are per-batch prefixes of input_pts
#define NBATCH 8

// ---------------------------------------------------------------------------
// KNN: one thread per query, LDS tile of source points, register top-K.
// ---------------------------------------------------------------------------
template<int K>
__global__ void knn_kernel(const float* __restrict__ pts, int srcN, int Mq,
                           int* __restrict__ idx)
{
    const int b = blockIdx.y;
    const int m = blockIdx.x * blockDim.x + threadIdx.x;
    const float* pb = pts + (size_t)b * PTS_STRIDE * 3;
    const bool act = (m < Mq);
    float qx = 0.f, qy = 0.f, qz = 0.f;
    if (act) { qx = pb[m*3+0]; qy = pb[m*3+1]; qz = pb[m*3+2]; }

    float bd[K]; int bi[K];
#pragma unroll
    for (int j = 0; j < K; ++j) { bd[j] = 3.4e38f; bi[j] = 0; }
    float worst = 3.4e38f; int wslot = 0;

    __shared__ float sp[128 * 3];
    for (int base = 0; base < srcN; base += 128) {
        const int cnt = min(128, srcN - base);
        __syncthreads();
        for (int i = threadIdx.x; i < cnt * 3; i += blockDim.x)
            sp[i] = pb[base*3 + i];
        __syncthreads();
        if (act) {
            for (int j = 0; j < cnt; ++j) {
                const float dx = sp[j*3+0]-qx, dy = sp[j*3+1]-qy, dz = sp[j*3+2]-qz;
                const float d = dx*dx + dy*dy + dz*dz;
                if (d < worst) {
                    bd[wslot] = d; bi[wslot] = base + j;
                    worst = -1.f;
#pragma unroll
                    for (int t = 0; t < K; ++t)
                        if (bd[t] > worst) { worst = bd[t]; wslot = t; }
                }
            }
        }
    }
    if (act) {
#pragma unroll
        for (int j = 0; j < K; ++j)
            idx[(size_t)(b*Mq + m)*K + j] = bi[j];
    }
}

// ---------------------------------------------------------------------------
// Fused gather + neighborhood normalize + weighting MLP + einsum.
// One block per query point. Writes F row [Cin*16] scaled by 1/K.
// ---------------------------------------------------------------------------
template<int K>
__global__ void ptconv_front_kernel(const float* __restrict__ xin,
                                    const float* __restrict__ pts,
                                    const int* __restrict__ idx,
                                    const float* __restrict__ cen,
                                    const float* __restrict__ l1w, const float* __restrict__ l1b,
                                    const float* __restrict__ l2w, const float* __restrict__ l2b,
                                    const float* __restrict__ l3w, const float* __restrict__ l3b,
                                    float* __restrict__ Fout,
                                    int rowBase, int Mq, int Cin, int srcN)
{
    const int tid = threadIdx.x;
    const int r = rowBase + blockIdx.x;
    const int b = r / Mq;
    const int mq = r - b * Mq;

    __shared__ int   sIdx[16];
    __shared__ float sRel[16*3];
    __shared__ float sD2[16];
    __shared__ float sInv;
    __shared__ float sFeat[768];    // [K][Cin], K*Cin <= 768 for all layers
    __shared__ float sD[16*48];
    __shared__ float sH1[16*32];
    __shared__ float sH2[16*16];
    __shared__ float sH3[16*16];

    const float* pb = pts + (size_t)b * PTS_STRIDE * 3;
    if (tid < K) sIdx[tid] = idx[(size_t)r * K + tid];
    __syncthreads();

    const float qx = pb[mq*3+0], qy = pb[mq*3+1], qz = pb[mq*3+2];
    if (tid < K) {
        const int j = sIdx[tid];
        const float rx = pb[j*3+0]-qx, ry = pb[j*3+1]-qy, rz = pb[j*3+2]-qz;
        sRel[tid*3+0] = rx; sRel[tid*3+1] = ry; sRel[tid*3+2] = rz;
        sD2[tid] = rx*rx + ry*ry + rz*rz;
    }
    __syncthreads();
    if (tid == 0) {
        float mx = 0.f;
#pragma unroll
        for (int k2 = 0; k2 < K; ++k2) mx = fmaxf(mx, sD2[k2]);
        const float mi = sqrtf(mx);
        sInv = (mi == 0.f) ? 1.f : (1.f / mi);
    }
    __syncthreads();

    const float inv = sInv;
    for (int i = tid; i < K*Cin; i += 128) {
        const int k2 = i / Cin, c = i - k2*Cin;
        sFeat[i] = xin[((size_t)b*srcN + sIdx[k2])*Cin + c];
    }
    for (int i = tid; i < K*48; i += 128) {
        const int k2 = i / 48, j = i - k2*48;
        sD[i] = sRel[k2*3 + (j >> 4)] * inv - cen[j];
    }
    __syncthreads();

    for (int i = tid; i < K*32; i += 128) {
        const int k2 = i >> 5, j = i & 31;
        float a = l1b[j];
        const float* w = l1w + j*48;
        const float* dd = sD + k2*48;
#pragma unroll
        for (int t = 0; t < 48; ++t) a += dd[t]*w[t];
        sH1[i] = a > 0.f ? a : 0.f;
    }
    __syncthreads();
    for (int i = tid; i < K*16; i += 128) {
        const int k2 = i >> 4, j = i & 15;
        float a = l2b[j];
        const float* w = l2w + j*32;
        const float* hh = sH1 + k2*32;
#pragma unroll
        for (int t = 0; t < 32; ++t) a += hh[t]*w[t];
        sH2[i] = a > 0.f ? a : 0.f;
    }
    __syncthreads();
    for (int i = tid; i < K*16; i += 128) {
        const int k2 = i >> 4, j = i & 15;
        float a = l3b[j];
        const float* w = l3w + j*16;
        const float* hh = sH2 + k2*16;
#pragma unroll
        for (int t = 0; t < 16; ++t) a += hh[t]*w[t];
        sH3[i] = a > 0.f ? a : 0.f;
    }
    __syncthreads();

    const float invK = 1.f / (float)K;
    const int CinNC = Cin * 16;
    float* fo = Fout + (size_t)blockIdx.x * CinNC;
    for (int i = tid; i < CinNC; i += 128) {
        const int cin = i >> 4, nc = i & 15;
        float a = 0.f;
#pragma unroll
        for (int k2 = 0; k2 < K; ++k2) a += sFeat[k2*Cin + cin] * sH3[k2*16 + nc];
        fo[i] = a * invK;
    }
}

// ---------------------------------------------------------------------------
// Dense mix GEMM via fp32 WMMA: out[R,Cout] = F[R,Kd] @ W[Kd,Cout].
// One wave per 16x16 output tile. F32 A 16x4 layout: lane-half = (K>=2),
// vgpr = K&1; B symmetric; C/D: vgpr r holds rows (r, r+8) per lane half.
// ---------------------------------------------------------------------------
__global__ void wmma_gemm_kernel(const float* __restrict__ A,
                                 const float* __restrict__ Bw,
                                 float* __restrict__ out,
                                 int Kd, int Cout, int outRowBase)
{
    const int m0 = blockIdx.x * 16;
    const int n0 = blockIdx.y * 16;
    const int lane = threadIdx.x;
    const int half = lane >> 4;
    const int lr = lane & 15;

    v8f c = {0.f,0.f,0.f,0.f,0.f,0.f,0.f,0.f};
    const float* arow = A + (size_t)(m0 + lr) * Kd + 2*half;
    const float* bcol = Bw + (size_t)(2*half) * Cout + n0 + lr;

    for (int kk = 0; kk < Kd; kk += 4) {
        v2f a, bf;
        a.x  = arow[kk];
        a.y  = arow[kk + 1];
        bf.x = bcol[(size_t)kk * Cout];
        bf.y = bcol[(size_t)(kk + 1) * Cout];
        c = __builtin_amdgcn_wmma_f32_16x16x4_f32(false, a, false, bf,
                                                  (short)0, c, false, false);
    }
#pragma unroll
    for (int r = 0; r < 8; ++r)
        out[(size_t)(outRowBase + m0 + r + 8*half) * Cout + n0 + lr] = c[r];
}

// ---------------------------------------------------------------------------
// Training-mode BatchNorm over all rows + ReLU. One block per channel.
// Writes into (possibly concat-strided) destination.
// ---------------------------------------------------------------------------
__global__ void bn_relu_kernel(const float* __restrict__ in, float* __restrict__ dst,
                               int R, int C, int dstStride, int dstCol0,
                               const float* __restrict__ g, const float* __restrict__ bta)
{
    const int c = blockIdx.x;
    const int tid = threadIdx.x;
    float s = 0.f, s2 = 0.f;
    for (int r = tid; r < R; r += blockDim.x) {
        const float v = in[(size_t)r*C + c];
        s += v; s2 += v*v;
    }
    __shared__ float rs[256], rq[256];
    rs[tid] = s; rq[tid] = s2;
    __syncthreads();
    for (int o = 128; o > 0; o >>= 1) {
        if (tid < o) { rs[tid] += rs[tid+o]; rq[tid] += rq[tid+o]; }
        __syncthreads();
    }
    __shared__ float sScale, sShift;
    if (tid == 0) {
        const float mu  = rs[0] / (float)R;
        const float var = rq[0] / (float)R - mu*mu;
        const float rsg = rsqrtf(var + 1e-5f);
        const float sc  = g[c] * rsg;
        sScale = sc; sShift = bta[c] - mu*sc;
    }
    __syncthreads();
    const float sc = sScale, sh = sShift;
    for (int r = tid; r < R; r += blockDim.x) {
        const float v = in[(size_t)r*C + c]*sc + sh;
        dst[(size_t)r*dstStride + dstCol0 + c] = v > 0.f ? v : 0.f;
    }
}

__global__ void copy_cols_kernel(float* __restrict__ dst, int dstStride, int col0,
                                 const float* __restrict__ src, int C, int R)
{
    const int i = blockIdx.x*blockDim.x + threadIdx.x;
    if (i >= R*C) return;
    const int r = i / C, c = i - r*C;
    dst[(size_t)r*dstStride + col0 + c] = src[i];
}

__global__ void fc_kernel(const float* __restrict__ xin, const float* __restrict__ w,
                          const float* __restrict__ bias, float* __restrict__ out, int R)
{
    const int i = blockIdx.x*blockDim.x + threadIdx.x;
    if (i >= R*13) return;
    const int r = i / 13, c = i - r*13;
    float a = bias[c];
    const float* xr = xin + (size_t)r*48;
    const float* wr = w + c*48;
#pragma unroll
    for (int t = 0; t < 48; ++t) a += xr[t]*wr[t];
    out[i] = a;
}

// ---------------------------------------------------------------------------
extern "C" void kernel_launch(void* const* d_in, const int* in_sizes, int n_in,
                              void* d_out, int out_size, void* d_ws, size_t ws_size,
                              hipStream_t stream)
{
    (void)in_sizes; (void)n_in; (void)out_size; (void)ws_size;
    const float* x   = (const float*)d_in[0];
    const float* pts = (const float*)d_in[1];

    struct CvP { const float *w, *cen, *l1w, *l1b, *l2w, *l2b, *l3w, *l3b; };
    struct BnP { const float *g, *b; };
    CvP cv[12]; BnP bn[12];
    int p = 2;
    for (int i = 0; i < 12; ++i) {
        cv[i].w   = (const float*)d_in[p++]; cv[i].cen = (const float*)d_in[p++];
        cv[i].l1w = (const float*)d_in[p++]; cv[i].l1b = (const float*)d_in[p++];
        cv[i].l2w = (const float*)d_in[p++]; cv[i].l2b = (const float*)d_in[p++];
        cv[i].l3w = (const float*)d_in[p++]; cv[i].l3b = (const float*)d_in[p++];
    }
    for (int i = 0; i < 12; ++i) {
        bn[i].g = (const float*)d_in[p++]; bn[i].b = (const float*)d_in[p++];
    }
    const float* fcw = (const float*)d_in[p++];
    const float* fcb = (const float*)d_in[p++];

    // --- workspace carve ---
    char* base = (char*)d_ws;
    auto alloc = [&](size_t nfloats) -> float* {
        float* r = (float*)base;
        base += ((nfloats*sizeof(float) + 255) / 256) * 256;
        return r;
    };
    float* x1   = alloc(16384*48);
    float* x2   = alloc(8192*48);
    float* x3   = alloc(2048*48);
    float* x4   = alloc(512*96);
    float* x5   = alloc(128*96);
    float* x6   = alloc(64*96);
    float* x5dc = alloc(128*192);
    float* x4dc = alloc(512*192);
    float* x3dc = alloc(2048*96);
    float* x2dc = alloc(8192*96);
    float* x1dc = alloc(16384*96);
    float* x0d  = alloc(65536*48);
    float* go   = alloc(65536*48);          // pre-BN GEMM output (max)
    float* Fbuf = alloc(6291456);           // chunked einsum staging (max 25MB)
    int*   idxb = (int*)alloc(524288);      // KNN indices (max)

    const int CHUNK = 4096;

    auto run = [&](int li, const float* xin, int Cin, int srcN, int Mq, int K,
                   int Cout, float* dst, int dstStride) {
        const int R = NBATCH * Mq;
        dim3 kg((Mq + 127) / 128, NBATCH);
        switch (K) {
            case 4:  knn_kernel<4> <<<kg, 128, 0, stream>>>(pts, srcN, Mq, idxb); break;
            case 8:  knn_kernel<8> <<<kg, 128, 0, stream>>>(pts, srcN, Mq, idxb); break;
            default: knn_kernel<16><<<kg, 128, 0, stream>>>(pts, srcN, Mq, idxb); break;
        }
        const int CinNC = Cin * 16;
        for (int r0 = 0; r0 < R; r0 += CHUNK) {
            const int nr = (R - r0 < CHUNK) ? (R - r0) : CHUNK;
            switch (K) {
                case 4:  ptconv_front_kernel<4> <<<nr, 128, 0, stream>>>(xin, pts, idxb,
                            cv[li].cen, cv[li].l1w, cv[li].l1b, cv[li].l2w, cv[li].l2b,
                            cv[li].l3w, cv[li].l3b, Fbuf, r0, Mq, Cin, srcN); break;
                case 8:  ptconv_front_kernel<8> <<<nr, 128, 0, stream>>>(xin, pts, idxb,
                            cv[li].cen, cv[li].l1w, cv[li].l1b, cv[li].l2w, cv[li].l2b,
                            cv[li].l3w, cv[li].l3b, Fbuf, r0, Mq, Cin, srcN); break;
                default: ptconv_front_kernel<16><<<nr, 128, 0, stream>>>(xin, pts, idxb,
                            cv[li].cen, cv[li].l1w, cv[li].l1b, cv[li].l2w, cv[li].l2b,
                            cv[li].l3w, cv[li].l3b, Fbuf, r0, Mq, Cin, srcN); break;
            }
            dim3 gg(nr / 16, Cout / 16);
            wmma_gemm_kernel<<<gg, 32, 0, stream>>>(Fbuf, cv[li].w, go, CinNC, Cout, r0);
        }
        bn_relu_kernel<<<Cout, 256, 0, stream>>>(go, dst, R, Cout, dstStride, 0,
                                                 bn[li].g, bn[li].b);
    };

    auto cat = [&](float* dst, int dstStride, int col0, const float* src, int C, int R) {
        copy_cols_kernel<<<(R*C + 255)/256, 256, 0, stream>>>(dst, dstStride, col0, src, C, R);
    };

    // encoder
    run(0,  x,    3,  8192, 2048, 16, 48, x1,  48);
    run(1,  x1,   48, 2048, 1024, 16, 48, x2,  48);
    run(2,  x2,   48, 1024, 256,  16, 48, x3,  48);
    run(3,  x3,   48, 256,  64,   8,  96, x4,  96);
    run(4,  x4,   96, 64,   16,   8,  96, x5,  96);
    run(5,  x5,   96, 16,   8,    4,  96, x6,  96);
    // decoder with skip concats
    run(6,  x6,   96,  8,   16,   4,  96, x5dc, 192);
    cat(x5dc, 192, 96, x5, 96, 128);
    run(7,  x5dc, 192, 16,  64,   4,  96, x4dc, 192);
    cat(x4dc, 192, 96, x4, 96, 512);
    run(8,  x4dc, 192, 64,  256,  4,  48, x3dc, 96);
    cat(x3dc, 96, 48, x3, 48, 2048);
    run(9,  x3dc, 96,  256, 1024, 8,  48, x2dc, 96);
    cat(x2dc, 96, 48, x2, 48, 8192);
    run(10, x2dc, 96,  1024, 2048, 8, 48, x1dc, 96);
    cat(x1dc, 96, 48, x1, 48, 16384);
    run(11, x1dc, 96,  2048, 8192, 8, 48, x0d, 48);

    fc_kernel<<<(65536*13 + 255)/256, 256, 0, stream>>>(x0d, fcw, fcb,
                                                        (float*)d_out, 65536);
}